// ModelWithNodeConcat_76484777607335
// MI455X (gfx1250) — compile-verified
//
#include <hip/hip_runtime.h>
#include <hip/hip_bf16.h>

#define N_NODES   100000
#define N_EDGES   1600000
#define N_GRAPHS  100
#define NPG       1000
#define HIDDEN    64
#define C_EDGE    16
#define MLP_HID   256
#define N_CLASSES 4
#define BN_SCALE  0.9999950000374997f   // 1/sqrt(1+1e-5)

typedef __attribute__((ext_vector_type(2))) float v2f;
typedef __attribute__((ext_vector_type(8))) float v8f;

__device__ __forceinline__ v8f wmma4(v2f a, v2f b, v8f c) {
    // V_WMMA_F32_16X16X4_F32:  D = A(16x4) x B(4x16) + C(16x16), all fp32
    return __builtin_amdgcn_wmma_f32_16x16x4_f32(
        /*neg_a=*/false, a, /*neg_b=*/false, b,
        /*c_mod=*/(short)0, c, /*reuse_a=*/false, /*reuse_b=*/false);
}

// ---------------------------------------------------------------- init: deg=1, EA=ones (self-loop term)
__global__ void k_init_nodes(float* __restrict__ deg, float* __restrict__ EA) {
    int t = blockIdx.x * blockDim.x + threadIdx.x;       // N_NODES*16 threads
    if (t >= N_NODES * C_EDGE) return;
    EA[t] = 1.0f;
    if ((t & 15) == 0) deg[t >> 4] = 1.0f;
}

// ---------------------------------------------------------------- once: deg / EA scatter over edges
__global__ void k_edge_deg_ea(const int* __restrict__ ei, const float* __restrict__ eattr,
                              float* __restrict__ deg, float* __restrict__ EA) {
    int t = blockIdx.x * blockDim.x + threadIdx.x;       // N_EDGES*16 threads
    if (t >= N_EDGES * C_EDGE) return;
    int e = t >> 4, c = t & 15;
    int dst = ei[N_EDGES + e];
    atomicAdd(&EA[dst * C_EDGE + c], eattr[e * C_EDGE + c]);
    if (c == 0) atomicAdd(&deg[dst], 1.0f);
}

// ---------------------------------------------------------------- per layer: S = x (self loop)
__global__ void k_copy_S(float* __restrict__ S, const float* __restrict__ x) {
    int t = blockIdx.x * blockDim.x + threadIdx.x;       // N_NODES*64 threads
    if (t < N_NODES * HIDDEN) S[t] = x[t];
}

// ---------------------------------------------------------------- per layer: S[dst] += x[src]
__global__ void k_edge_scatter(const int* __restrict__ ei, const float* __restrict__ x,
                               float* __restrict__ S) {
    int t = blockIdx.x * blockDim.x + threadIdx.x;       // N_EDGES*64 threads
    if (t >= N_EDGES * HIDDEN) return;
    int e = t >> 6, c = t & 63;
    int src = ei[e];
    int dst = ei[N_EDGES + e];
    atomicAdd(&S[dst * HIDDEN + c], x[src * HIDDEN + c]);
}

// ---------------------------------------------------------------- per layer: fused WMMA node GEMM + bias + relu + BN + relu
// agg = deg.*(x@Wdst) + S@Wsrc + EA@We + deg.*b ; xnext = relu(g*relu(agg)*s + beta)
// grid = N/16 blocks, blockDim = 128 (4 waves; wave nb handles output cols nb*16..nb*16+15)
__global__ void k_layer_wmma(const float* __restrict__ xprev, const float* __restrict__ S,
                             const float* __restrict__ EA,    const float* __restrict__ deg,
                             const float* __restrict__ Wl,    // [144,64] row-major
                             const float* __restrict__ bl, const float* __restrict__ gl,
                             const float* __restrict__ betal, float* __restrict__ xnext) {
    const int lane = threadIdx.x & 31;
    const int nb   = threadIdx.x >> 5;            // 0..3 (col block)
    const int mb   = blockIdx.x;                  // row block (16 nodes)
    const int mrow = lane & 15;
    const int hi   = lane >> 4;                   // 0 | 1
    const int kh   = hi << 1;                     // K offset: 0 | 2
    const int node = mb * 16 + mrow;              // A-fragment row
    const int col  = nb * 16 + mrow;              // B / C-fragment column
    const float d  = deg[node];

    const float* xrow  = xprev + node * HIDDEN;
    const float* srow  = S     + node * HIDDEN;
    const float* earow = EA    + node * C_EDGE;

    v8f acc = {0.f, 0.f, 0.f, 0.f, 0.f, 0.f, 0.f, 0.f};

    // term 1: (deg .* x) @ W[0:64]   (deg folded into A rows)
#pragma unroll
    for (int k = 0; k < 64; k += 4) {
        v2f a, b;
        a.x = d * xrow[k + kh];
        a.y = d * xrow[k + kh + 1];
        b.x = Wl[(k + kh)     * HIDDEN + col];
        b.y = Wl[(k + kh + 1) * HIDDEN + col];
        acc = wmma4(a, b, acc);
    }
    // term 2: S @ W[64:128]
#pragma unroll
    for (int k = 0; k < 64; k += 4) {
        v2f a, b;
        a.x = srow[k + kh];
        a.y = srow[k + kh + 1];
        b.x = Wl[(64 + k + kh)     * HIDDEN + col];
        b.y = Wl[(64 + k + kh + 1) * HIDDEN + col];
        acc = wmma4(a, b, acc);
    }
    // term 3: EA @ W[128:144]
#pragma unroll
    for (int k = 0; k < 16; k += 4) {
        v2f a, b;
        a.x = earow[k + kh];
        a.y = earow[k + kh + 1];
        b.x = Wl[(128 + k + kh)     * HIDDEN + col];
        b.y = Wl[(128 + k + kh + 1) * HIDDEN + col];
        acc = wmma4(a, b, acc);
    }

    const float bc = bl[col], gc = gl[col], bb = betal[col];
#pragma unroll
    for (int v = 0; v < 8; ++v) {                 // C/D layout: row = v + 8*hi, col = lane&15
        int cnode = mb * 16 + v + (hi << 3);
        float agg = acc[v] + deg[cnode] * bc;
        float h   = fmaxf(agg, 0.0f);
        float o   = fmaxf(gc * h * BN_SCALE + bb, 0.0f);
        xnext[cnode * HIDDEN + col] = o;
    }
}

// ---------------------------------------------------------------- per layer: emb_l -> feats[:, 64+64l : 128+64l]
__global__ void k_gather_emb(const float* __restrict__ x, const int* __restrict__ spec,
                             float* __restrict__ feats, int l) {
    int t = blockIdx.x * blockDim.x + threadIdx.x;       // 100*64 threads
    if (t >= N_GRAPHS * HIDDEN) return;
    int g = t >> 6, c = t & 63;
    feats[g * 320 + 64 + l * 64 + c] = x[spec[g] * HIDDEN + c];
}

// ---------------------------------------------------------------- pooled -> feats[:, 0:64]
__global__ void k_pool(const float* __restrict__ x, float* __restrict__ feats) {
    int g = blockIdx.x;                                  // 100 blocks x 64 threads
    int c = threadIdx.x;
    float s = 0.0f;
    const float* base = x + (size_t)g * NPG * HIDDEN + c;
    for (int n = 0; n < NPG; ++n) s += base[n * HIDDEN];
    feats[g * 320 + c] = s;
}

// ---------------------------------------------------------------- neighbor -> feats[:, 256:320]
__global__ void k_neighbor(const float* __restrict__ nb, float* __restrict__ feats) {
    int t = blockIdx.x * blockDim.x + threadIdx.x;       // 100*64 threads
    if (t >= N_GRAPHS * HIDDEN) return;
    int g = t >> 6, c = t & 63;
    feats[g * 320 + 256 + c] = nb[g * HIDDEN + c];
}

// ---------------------------------------------------------------- head: h1 = relu(feats @ fc1 + b1)
__global__ void k_fc1(const float* __restrict__ feats, const float* __restrict__ w,
                      const float* __restrict__ b, float* __restrict__ h1) {
    int g = blockIdx.x, j = threadIdx.x;                 // 100 blocks x 256 threads
    float acc = b[j];
    const float* fr = feats + g * 320;
    for (int r = 0; r < 320; ++r) acc += fr[r] * w[r * MLP_HID + j];
    h1[g * MLP_HID + j] = fmaxf(acc, 0.0f);
}

// ---------------------------------------------------------------- head: out = h1 @ fc2 + b2
__global__ void k_fc2(const float* __restrict__ h1, const float* __restrict__ w,
                      const float* __restrict__ b, float* __restrict__ out) {
    int t = blockIdx.x * blockDim.x + threadIdx.x;       // 100*4 threads
    if (t >= N_GRAPHS * N_CLASSES) return;
    int g = t >> 2, k = t & 3;
    float acc = b[k];
    const float* hr = h1 + g * MLP_HID;
    for (int j = 0; j < MLP_HID; ++j) acc += hr[j] * w[j * N_CLASSES + k];
    out[g * N_CLASSES + k] = acc;
}

extern "C" void kernel_launch(void* const* d_in, const int* in_sizes, int n_in,
                              void* d_out, int out_size, void* d_ws, size_t ws_size,
                              hipStream_t stream) {
    const float* x      = (const float*)d_in[0];
    const float* eattr  = (const float*)d_in[1];
    const float* neigh  = (const float*)d_in[2];
    const float* W      = (const float*)d_in[3];   // [3,144,64]
    const float* b      = (const float*)d_in[4];   // [3,64]
    const float* gamma  = (const float*)d_in[5];
    const float* beta   = (const float*)d_in[6];
    const float* fc1w   = (const float*)d_in[7];   // [320,256]
    const float* fc1b   = (const float*)d_in[8];
    const float* fc2w   = (const float*)d_in[9];   // [256,4]
    const float* fc2b   = (const float*)d_in[10];
    const int*   ei     = (const int*)d_in[11];    // [2,E]
    const int*   spec   = (const int*)d_in[13];
    float*       out    = (float*)d_out;

    float* ws   = (float*)d_ws;
    float* deg  = ws;                              // N
    float* EA   = deg + N_NODES;                   // N*16
    float* S    = EA  + (size_t)N_NODES * 16;      // N*64
    float* xA   = S   + (size_t)N_NODES * 64;      // N*64
    float* xB   = xA  + (size_t)N_NODES * 64;      // N*64
    float* feats= xB  + (size_t)N_NODES * 64;      // 100*320
    float* h1   = feats + N_GRAPHS * 320;          // 100*256

    // layer-invariant: degree + edge-attr aggregation (incl. self-loop ones)
    k_init_nodes <<<(N_NODES * 16 + 255) / 256, 256, 0, stream>>>(deg, EA);
    k_edge_deg_ea<<<(N_EDGES * 16 + 255) / 256, 256, 0, stream>>>(ei, eattr, deg, EA);

    const float* xprev = x;
    float* bufs[3] = {xA, xB, xA};
    for (int l = 0; l < 3; ++l) {
        float* xnext = bufs[l];
        k_copy_S      <<<(N_NODES * 64 + 255) / 256, 256, 0, stream>>>(S, xprev);
        k_edge_scatter<<<(N_EDGES * 64 + 255) / 256, 256, 0, stream>>>(ei, xprev, S);
        k_layer_wmma  <<<N_NODES / 16, 128, 0, stream>>>(
            xprev, S, EA, deg, W + (size_t)l * 144 * 64,
            b + l * 64, gamma + l * 64, beta + l * 64, xnext);
        k_gather_emb  <<<(N_GRAPHS * 64 + 63) / 64, 64, 0, stream>>>(xnext, spec, feats, l);
        xprev = xnext;
    }

    k_pool    <<<N_GRAPHS, 64, 0, stream>>>(xprev, feats);
    k_neighbor<<<(N_GRAPHS * 64 + 255) / 256, 256, 0, stream>>>(neigh, feats);
    k_fc1     <<<N_GRAPHS, MLP_HID, 0, stream>>>(feats, fc1w, fc1b, h1);
    k_fc2     <<<(N_GRAPHS * N_CLASSES + 63) / 64, 64, 0, stream>>>(h1, fc2w, fc2b, out);
}